// T2RCell_68659347194398
// MI455X (gfx1250) — compile-verified
//
#include <hip/hip_runtime.h>
#include <math.h>

// ---------------- problem constants ----------------
#define B_   16
#define C_   384
#define N_   4096          // H*W
#define NH_  4
#define HC_  96            // C/NH
#define BCN  ((size_t)B_ * C_ * N_)   // 25,165,824 floats per tensor

typedef float v2f __attribute__((ext_vector_type(2)));
typedef float v8f __attribute__((ext_vector_type(8)));

// D(16x16,f32) = A(16x4,f32) x B(4x16,f32) + C  — CDNA5 V_WMMA_F32_16X16X4_F32
__device__ __forceinline__ v8f wmma4(v2f a, v2f b, v8f c) {
    return __builtin_amdgcn_wmma_f32_16x16x4_f32(
        /*neg_a=*/false, a, /*neg_b=*/false, b,
        /*c_mod=*/(short)0, c, /*reuse_a=*/false, /*reuse_b=*/false);
}

// ------------- 1) spatial mean/max pool for router -------------
// grid = B*2C blocks, 256 threads; f row layout: [max(2C) | avg(2C)] = 1536
__global__ void pool_kernel(const float* __restrict__ x1,
                            const float* __restrict__ x2,
                            float* __restrict__ f) {
    __shared__ float smax[256];
    __shared__ float ssum[256];
    const int bc = blockIdx.x;
    const int b  = bc / (2 * C_);
    const int ch = bc % (2 * C_);
    const float* src = (ch < C_) ? (x1 + ((size_t)b * C_ + ch) * N_)
                                 : (x2 + ((size_t)b * C_ + (ch - C_)) * N_);
    float m = -__builtin_inff(), s = 0.f;
    for (int i = threadIdx.x; i < N_; i += 256) {
        float v = src[i];
        m = fmaxf(m, v);
        s += v;
    }
    smax[threadIdx.x] = m;
    ssum[threadIdx.x] = s;
    __syncthreads();
    for (int st = 128; st > 0; st >>= 1) {
        if (threadIdx.x < st) {
            smax[threadIdx.x] = fmaxf(smax[threadIdx.x], smax[threadIdx.x + st]);
            ssum[threadIdx.x] += ssum[threadIdx.x + st];
        }
        __syncthreads();
    }
    if (threadIdx.x == 0) {
        f[(size_t)b * (4 * C_) + ch]            = smax[0];
        f[(size_t)b * (4 * C_) + 2 * C_ + ch]   = ssum[0] * (1.0f / N_);
    }
}

// ------------- 2) router MLP: relu(f@W1^T+b1) -> relu(tanh(@W2^T+b2)) -------------
// grid = B blocks, 256 threads
__global__ void router_mlp(const float* __restrict__ f,
                           const float* __restrict__ W1, const float* __restrict__ b1,
                           const float* __restrict__ W2, const float* __restrict__ b2,
                           float* __restrict__ path) {
    __shared__ float fs[1536];
    __shared__ float hs[768];
    const int b = blockIdx.x;
    for (int i = threadIdx.x; i < 1536; i += 256) fs[i] = f[(size_t)b * 1536 + i];
    __syncthreads();
    for (int e = threadIdx.x; e < 768; e += 256) {
        const float* wr = W1 + (size_t)e * 1536;
        float s = b1[e];
        for (int i = 0; i < 1536; ++i) s += wr[i] * fs[i];
        hs[e] = fmaxf(s, 0.f);
    }
    __syncthreads();
    if (threadIdx.x < 4) {
        const float* wr = W2 + (size_t)threadIdx.x * 768;
        float s = b2[threadIdx.x];
        for (int i = 0; i < 768; ++i) s += wr[i] * hs[i];
        path[b * 4 + threadIdx.x] = fmaxf(tanhf(s), 0.f);
    }
}

// ------------- 3) 1x1-conv GEMM: Y[b,o,n] = sum_c W[o,c] X[b,c,n] (+Res) -------------
// one 16x16 output tile per wave; 4 waves/block; grid = B*(C/16)*(N/16)/4 = 24576
__global__ void proj_gemm(const float* __restrict__ W, const float* __restrict__ X,
                          const float* __restrict__ Res, float* __restrict__ Y) {
    const int lane = threadIdx.x & 31;
    const int w    = blockIdx.x * 4 + (threadIdx.x >> 5);
    const int b    = w / ((C_ / 16) * (N_ / 16));
    const int r    = w % ((C_ / 16) * (N_ / 16));
    const int mt   = r / (N_ / 16);
    const int nt   = r % (N_ / 16);
    const int l15  = lane & 15;
    const int subk = (lane >> 4) << 1;           // 0 or 2

    const float* Wrow = W + (size_t)(mt * 16 + l15) * C_;
    const float* Xb   = X + (size_t)b * C_ * N_;
    const int ncol    = nt * 16 + l15;

    v8f acc = {};
    for (int kk = 0; kk < C_; kk += 4) {
        v2f a = *(const v2f*)(Wrow + kk + subk);            // A: 16x4 tile of W
        v2f bb;
        bb.x = Xb[(size_t)(kk + subk) * N_ + ncol];         // B: 4x16 tile of X
        bb.y = Xb[(size_t)(kk + subk + 1) * N_ + ncol];
        acc = wmma4(a, bb, acc);
    }

    const int mbase = mt * 16 + ((lane >> 4) << 3);
    float* Yb = Y + (size_t)b * C_ * N_;
    const float* Rb = Res ? (Res + (size_t)b * C_ * N_) : nullptr;
#pragma unroll
    for (int rr = 0; rr < 8; ++rr) {
        size_t idx = (size_t)(mbase + rr) * N_ + ncol;
        float v = acc[rr];
        if (Rb) v += Rb[idx];
        Yb[idx] = v;
    }
}

// ------------- 4) reciprocal row L2-norms for q and k -------------
// grid = 2*B*C blocks (q rows then k rows), 256 threads
__global__ void norm_kernel(const float* __restrict__ q, const float* __restrict__ k,
                            float* __restrict__ inv) {
    __shared__ float ssum[256];
    const int idx = blockIdx.x;
    const float* src = (idx < B_ * C_) ? (q + (size_t)idx * N_)
                                       : (k + (size_t)(idx - B_ * C_) * N_);
    float s = 0.f;
    for (int i = threadIdx.x; i < N_; i += 256) {
        float v = src[i];
        s += v * v;
    }
    ssum[threadIdx.x] = s;
    __syncthreads();
    for (int st = 128; st > 0; st >>= 1) {
        if (threadIdx.x < st) ssum[threadIdx.x] += ssum[threadIdx.x + st];
        __syncthreads();
    }
    if (threadIdx.x == 0) inv[idx] = 1.0f / fmaxf(sqrtf(ssum[0]), 1e-12f);
}

// ------------- 5) S[b,h,i,j] = (q_i . k_j) * invq_i * invk_j * temp[h] -------------
// M=N'=96 (6x6 tiles), K=4096; grid = B*NH*36/4 = 576 blocks of 128
__global__ void attn_qk(const float* __restrict__ q, const float* __restrict__ k,
                        const float* __restrict__ invq, const float* __restrict__ invk,
                        const float* __restrict__ temp, float* __restrict__ S) {
    const int lane = threadIdx.x & 31;
    const int w    = blockIdx.x * 4 + (threadIdx.x >> 5);
    const int bh   = w / 36;
    const int t    = w % 36;
    const int mt   = t / 6, nt = t % 6;
    const int b    = bh / NH_, h = bh % NH_;
    const int l15  = lane & 15;
    const int subk = (lane >> 4) << 1;

    const int rowc = h * HC_ + mt * 16 + l15;
    const int colc = h * HC_ + nt * 16 + l15;
    const float* qrow = q + ((size_t)b * C_ + rowc) * N_;
    const float* krow = k + ((size_t)b * C_ + colc) * N_;

    v8f acc = {};
    for (int kk = 0; kk < N_; kk += 4) {
        v2f a  = *(const v2f*)(qrow + kk + subk);
        v2f bb = *(const v2f*)(krow + kk + subk);
        acc = wmma4(a, bb, acc);
    }

    const float tm = temp[h];
    const float ik = invk[b * C_ + colc];
    const int mbase = mt * 16 + ((lane >> 4) << 3);
    float* Sb = S + (size_t)bh * HC_ * HC_;
#pragma unroll
    for (int rr = 0; rr < 8; ++rr) {
        const int i = mbase + rr;
        const float iq = invq[b * C_ + h * HC_ + i];
        Sb[(size_t)i * HC_ + nt * 16 + l15] = acc[rr] * iq * ik * tm;
    }
}

// ------------- 6) softmax over last dim (96) -------------
// grid = B*NH*96 rows, 96 threads
__global__ void softmax96(float* __restrict__ S) {
    __shared__ float buf[HC_];
    __shared__ float red[2];
    const int j = threadIdx.x;
    float* p = S + (size_t)blockIdx.x * HC_;
    float v = p[j];
    buf[j] = v;
    __syncthreads();
    if (j == 0) {
        float m = buf[0];
        for (int i = 1; i < HC_; ++i) m = fmaxf(m, buf[i]);
        red[0] = m;
    }
    __syncthreads();
    float e = expf(v - red[0]);
    buf[j] = e;
    __syncthreads();
    if (j == 0) {
        float s = 0.f;
        for (int i = 0; i < HC_; ++i) s += buf[i];
        red[1] = s;
    }
    __syncthreads();
    p[j] = e / red[1];
}

// ------------- 7) O[b,h,i,n] = sum_j P[i,j] * v[b,h*96+j,n] -------------
// M=96 (6 tiles), N=4096 (256 tiles), K=96; grid = B*NH*1536/4 = 24576
__global__ void attn_pv(const float* __restrict__ P, const float* __restrict__ v,
                        float* __restrict__ O) {
    const int lane = threadIdx.x & 31;
    const int w    = blockIdx.x * 4 + (threadIdx.x >> 5);
    const int bh   = w / (6 * (N_ / 16));
    const int t    = w % (6 * (N_ / 16));
    const int mt   = t / (N_ / 16), nt = t % (N_ / 16);
    const int b    = bh / NH_, h = bh % NH_;
    const int l15  = lane & 15;
    const int subk = (lane >> 4) << 1;

    const float* Prow = P + ((size_t)bh * HC_ + mt * 16 + l15) * HC_;
    const float* Vb   = v + ((size_t)b * C_ + h * HC_) * N_;
    const int ncol    = nt * 16 + l15;

    v8f acc = {};
    for (int kk = 0; kk < HC_; kk += 4) {
        v2f a = *(const v2f*)(Prow + kk + subk);
        v2f bb;
        bb.x = Vb[(size_t)(kk + subk) * N_ + ncol];
        bb.y = Vb[(size_t)(kk + subk + 1) * N_ + ncol];
        acc = wmma4(a, bb, acc);
    }
    const int mbase = mt * 16 + ((lane >> 4) << 3);
    float* Ob = O + ((size_t)b * C_ + h * HC_) * N_;
#pragma unroll
    for (int rr = 0; rr < 8; ++rr)
        Ob[(size_t)(mbase + rr) * N_ + ncol] = acc[rr];
}

// =====================================================================
extern "C" void kernel_launch(void* const* d_in, const int* in_sizes, int n_in,
                              void* d_out, int out_size, void* d_ws, size_t ws_size,
                              hipStream_t stream) {
    const float* x1  = (const float*)d_in[0];
    const float* x2  = (const float*)d_in[1];
    const float* Wq  = (const float*)d_in[2];
    const float* Wk  = (const float*)d_in[3];
    const float* Wv  = (const float*)d_in[4];
    const float* Wp  = (const float*)d_in[5];
    const float* tmp = (const float*)d_in[6];
    const float* W1  = (const float*)d_in[7];
    const float* b1  = (const float*)d_in[8];
    const float* W2  = (const float*)d_in[9];
    const float* b2  = (const float*)d_in[10];

    // workspace layout (floats)
    float* ws  = (float*)d_ws;
    float* q   = ws;                                   // BCN  (reused for O)
    float* k   = ws + BCN;                             // BCN
    float* v   = ws + 2 * BCN;                         // BCN
    float* S   = ws + 3 * BCN;                         // B*NH*96*96
    float* f   = S + (size_t)B_ * NH_ * HC_ * HC_;     // B*1536
    float* inv = f + (size_t)B_ * 4 * C_;              // 2*B*C (invq | invk)

    float* cross = (float*)d_out;            // [B,C,H,W]
    float* x2out = cross + BCN;              // [B,C,H,W]
    float* path  = cross + 2 * BCN;          // [B,4]

    // router
    pool_kernel<<<B_ * 2 * C_, 256, 0, stream>>>(x1, x2, f);
    router_mlp<<<B_, 256, 0, stream>>>(f, W1, b1, W2, b2, path);

    // q/k/v projections (WMMA GEMMs)
    const int projBlocks = (B_ * (C_ / 16) * (N_ / 16)) / 4;   // 24576
    proj_gemm<<<projBlocks, 128, 0, stream>>>(Wq, x1, nullptr, q);
    proj_gemm<<<projBlocks, 128, 0, stream>>>(Wk, x2, nullptr, k);
    proj_gemm<<<projBlocks, 128, 0, stream>>>(Wv, x2, nullptr, v);

    // row norms, S = q k^T (scaled), softmax, O = P v  (O overwrites q buffer)
    norm_kernel<<<2 * B_ * C_, 256, 0, stream>>>(q, k, inv);
    attn_qk<<<(B_ * NH_ * 36) / 4, 128, 0, stream>>>(q, k, inv, inv + B_ * C_, tmp, S);
    softmax96<<<B_ * NH_ * HC_, HC_, 0, stream>>>(S);
    attn_pv<<<(B_ * NH_ * 6 * (N_ / 16)) / 4, 128, 0, stream>>>(S, v, q);

    // final projection + residual into d_out
    proj_gemm<<<projBlocks, 128, 0, stream>>>(Wp, q, x1, cross);

    // second output: x2 passthrough
    hipMemcpyAsync(x2out, x2, BCN * sizeof(float), hipMemcpyDeviceToDevice, stream);
}